// MambaLayer_17824114278653
// MI455X (gfx1250) — compile-verified
//
#include <hip/hip_runtime.h>
#include <math.h>

// ---------------------------------------------------------------------------
// Mamba-style masked SSM scan, chunked (SSD) formulation for CDNA5 / gfx1250.
// T=4096, H=16, N=16, D=64. Chunk length 16 -> every matmul is a 16x16 tile
// mapped onto V_WMMA_F32_16X16X4_F32 (full fp32, matches reference precision).
// ---------------------------------------------------------------------------

typedef __attribute__((ext_vector_type(2))) float v2f;
typedef __attribute__((ext_vector_type(8))) float v8f;

#define T_LEN 4096
#define NH 16
#define NN 16
#define ND 64
#define CHUNK 16
#define KCH (T_LEN / CHUNK)        // 256 chunks
#define STATE (NN * ND)            // 1024 floats per (chunk,head) state
#define CARRY_OFF (NH * NN * ND)   // next_carry comes first in d_out

static __device__ __forceinline__ v8f wmma_f32x4(v2f a, v2f b, v8f c) {
  // D(16x16,f32) = A(16x4,f32) x B(4x16,f32) + C
  return __builtin_amdgcn_wmma_f32_16x16x4_f32(
      /*neg_a=*/false, a, /*neg_b=*/false, b,
      /*c_mod=*/(short)0, c, /*reuse_a=*/false, /*reuse_b=*/false);
}

// ---------------------------------------------------------------------------
// Pass 1: per (chunk,head) wave computes
//   Ach    = prod_{s=0..15} a'_s                      (chunk transition scalar)
//   Slocal = B_chunk^T @ diag(dt_s * suffixDecay_s) @ X_chunk   (16x64, WMMA)
// ---------------------------------------------------------------------------
__global__ __launch_bounds__(32)
void k1_chunk_local(const float* __restrict__ x, const float* __restrict__ dt,
                    const float* __restrict__ Bg, const float* __restrict__ mask,
                    const float* __restrict__ log_decay,
                    float* __restrict__ Sbuf, float* __restrict__ Ach) {
  const int k    = blockIdx.x;       // chunk
  const int h    = blockIdx.y;       // head
  const int lane = threadIdx.x;
  const int m    = lane & 15;
  const int half = lane >> 4;        // 0: K={0,1}, 1: K={2,3} per WMMA step
  const int t0   = k * CHUNK;

  const float rate = -expf(log_decay[h]);

  // a'_s (mask folds reset into the decay), E[s] = dt_s * prod_{u>s} a'_u
  float ap[CHUNK], E[CHUNK];
#pragma unroll
  for (int s = 0; s < CHUNK; ++s) {
    const float dts = dt[(t0 + s) * NH + h];
    const float a   = expf(dts * rate);
    ap[s] = (mask[t0 + s] != 0.0f) ? 0.0f : a;
    E[s]  = dts;
  }
  float suf = 1.0f;
#pragma unroll
  for (int s = CHUNK - 1; s >= 0; --s) {
    E[s] = E[s] * suf;   // dt_s * A(s, chunk_end)
    suf *= ap[s];
  }
  if (lane == 0) Ach[k * NH + h] = suf;   // full-chunk transition

  // Slocal(16n x 64d) = sum_s B[s][n] * E[s] * x[s][d], as 4 D-tiles of WMMA
#pragma unroll
  for (int dt4 = 0; dt4 < 4; ++dt4) {
    const int d = dt4 * 16 + m;
    v8f acc = {};
#pragma unroll
    for (int kk = 0; kk < 4; ++kk) {
      const int s0 = kk * 4 + half * 2;
      const int s1 = s0 + 1;
      v2f av, bv;
      // A-matrix = B^T : row n = m, K = s
      av.x = Bg[((t0 + s0) * NH + h) * NN + m];
      av.y = Bg[((t0 + s1) * NH + h) * NN + m];
      // B-matrix = E[s] * X : row s, col d = m
      const float e0 = half ? E[kk * 4 + 2] : E[kk * 4 + 0];
      const float e1 = half ? E[kk * 4 + 3] : E[kk * 4 + 1];
      bv.x = e0 * x[((t0 + s0) * NH + h) * ND + d];
      bv.y = e1 * x[((t0 + s1) * NH + h) * ND + d];
      acc = wmma_f32x4(av, bv, acc);
    }
#pragma unroll
    for (int r = 0; r < 8; ++r) {
      const int n = r + half * 8;
      Sbuf[(((size_t)k * NH + h) * NN + n) * ND + d] = acc[r];
    }
  }
}

// ---------------------------------------------------------------------------
// Pass 2: per head, scan 256 chunk states. Replaces Slocal[k] in-place with
// the INCOMING state of chunk k; emits the final carry to d_out[0:16384].
// ---------------------------------------------------------------------------
__global__ __launch_bounds__(1024)
void k2_scan(const float* __restrict__ init_carry, const float* __restrict__ Ach,
             float* __restrict__ Sbuf, float* __restrict__ out_carry) {
  const int h = blockIdx.x;
  const int e = threadIdx.x;          // flat (n*64 + d)
  float S = init_carry[h * STATE + e];
  for (int k = 0; k < KCH; ++k) {
    const size_t idx = ((size_t)k * NH + h) * STATE + e;
    const float sl = Sbuf[idx];
    const float a  = Ach[k * NH + h];
    Sbuf[idx] = S;                    // state entering chunk k
    S = a * S + sl;
  }
  out_carry[h * STATE + e] = S;
}

// ---------------------------------------------------------------------------
// Pass 3: per (chunk,head) wave computes the 16x64 output tile:
//   O = (W ⊙ (C@B^T)) @ X  +  (P_t ⊙ C) @ S_in  +  skip ⊙ X
// ---------------------------------------------------------------------------
__global__ __launch_bounds__(32)
void k3_output(const float* __restrict__ x, const float* __restrict__ dt,
               const float* __restrict__ Bg, const float* __restrict__ Cg,
               const float* __restrict__ mask, const float* __restrict__ log_decay,
               const float* __restrict__ skipw,
               const float* __restrict__ Sbuf, float* __restrict__ out) {
  __shared__ float Ms[16][17];        // padded: C/D-layout -> A-layout transit
  const int k    = blockIdx.x;
  const int h    = blockIdx.y;
  const int lane = threadIdx.x;
  const int m    = lane & 15;
  const int half = lane >> 4;
  const int t0   = k * CHUNK;

  const float rate = -expf(log_decay[h]);
  float ap[CHUNK];
#pragma unroll
  for (int s = 0; s < CHUNK; ++s) {
    const float dts = dt[(t0 + s) * NH + h];
    const float a   = expf(dts * rate);
    ap[s] = (mask[t0 + s] != 0.0f) ? 0.0f : a;
  }

  // G[t][s] = C_t · B_s  (A = C with K=n ; B-mat = B^T with col s = m)
  v8f g = {};
#pragma unroll
  for (int kk = 0; kk < 4; ++kk) {
    const int n0 = kk * 4 + half * 2;
    v2f av, bv;
    av.x = Cg[((t0 + m) * NH + h) * NN + n0];
    av.y = Cg[((t0 + m) * NH + h) * NN + n0 + 1];
    bv.x = Bg[((t0 + m) * NH + h) * NN + n0];
    bv.y = Bg[((t0 + m) * NH + h) * NN + n0 + 1];
    g = wmma_f32x4(av, bv, g);
  }

  // Causal weights W[t][s]: s = m (column of this lane), t = r + 8*half.
  //   t<s: 0 ; t==s: dt_s ; t>s: dt_s * prod_{u=s+1..t} a'_u
  const float dts = dt[(t0 + m) * NH + h];
  float w[8];
#pragma unroll
  for (int r = 0; r < 8; ++r) w[r] = 0.0f;
  {
    float run = dts;
#pragma unroll
    for (int t = 0; t < 16; ++t) {
      run = (t > m) ? run * ap[t] : run;
      const float val = (t < m) ? 0.0f : run;
      w[t & 7] = ((t >> 3) == half) ? val : w[t & 7];
    }
  }

  // M = W ⊙ G, stage through LDS to re-pack from C/D layout into A layout
#pragma unroll
  for (int r = 0; r < 8; ++r) Ms[r + 8 * half][m] = w[r] * g[r];
  __syncthreads();

  // P_t = prod_{u=0..t} a'_u for this lane's A-matrix row t = m
  float Pt = 1.0f;
#pragma unroll
  for (int u = 0; u < 16; ++u) Pt = (u <= m) ? Pt * ap[u] : Pt;

  const float* Sin = Sbuf + ((size_t)k * NH + h) * STATE;

#pragma unroll
  for (int dt4 = 0; dt4 < 4; ++dt4) {
    const int d = dt4 * 16 + m;
    v8f acc = {};
#pragma unroll
    for (int kk = 0; kk < 4; ++kk) {
      const int s0 = kk * 4 + half * 2;   // doubles as n0 for the state term
      // intra-chunk: M @ X
      v2f av, bv;
      av.x = Ms[m][s0];
      av.y = Ms[m][s0 + 1];
      bv.x = x[((t0 + s0) * NH + h) * ND + d];
      bv.y = x[((t0 + s0 + 1) * NH + h) * ND + d];
      acc = wmma_f32x4(av, bv, acc);
      // carried state: (P_t * C) @ S_in
      v2f av2, bv2;
      av2.x = Cg[((t0 + m) * NH + h) * NN + s0] * Pt;
      av2.y = Cg[((t0 + m) * NH + h) * NN + s0 + 1] * Pt;
      bv2.x = Sin[s0 * ND + d];
      bv2.y = Sin[(s0 + 1) * ND + d];
      acc = wmma_f32x4(av2, bv2, acc);
    }
    // epilogue: skip connection + store (D-layout: row t = r + 8*half)
#pragma unroll
    for (int r = 0; r < 8; ++r) {
      const int t = r + 8 * half;
      const float xv = x[((t0 + t) * NH + h) * ND + d];
      out[CARRY_OFF + (((size_t)(t0 + t)) * NH + h) * ND + d] =
          acc[r] + skipw[h * ND + d] * xv;
    }
  }
}

// ---------------------------------------------------------------------------
extern "C" void kernel_launch(void* const* d_in, const int* in_sizes, int n_in,
                              void* d_out, int out_size, void* d_ws, size_t ws_size,
                              hipStream_t stream) {
  const float* x    = (const float*)d_in[0];  // (T,H,D)
  const float* dt   = (const float*)d_in[1];  // (T,H)
  const float* B    = (const float*)d_in[2];  // (T,H,N)
  const float* C    = (const float*)d_in[3];  // (T,H,N)
  const float* mask = (const float*)d_in[4];  // (T,)
  const float* ic   = (const float*)d_in[5];  // (H,N,D)
  const float* ld   = (const float*)d_in[6];  // (H,)
  const float* sw   = (const float*)d_in[7];  // (H,D)
  float* out = (float*)d_out;                 // [carry(H,N,D) | output(T,H,D)]

  float* Sbuf = (float*)d_ws;                           // 256*16*1024 f32 = 16 MB
  float* Ach  = Sbuf + (size_t)KCH * NH * STATE;        // 4096 f32

  dim3 gridc(KCH, NH);
  k1_chunk_local<<<gridc, 32, 0, stream>>>(x, dt, B, mask, ld, Sbuf, Ach);
  k2_scan<<<NH, STATE, 0, stream>>>(ic, Ach, Sbuf, out);
  k3_output<<<gridc, 32, 0, stream>>>(x, dt, B, C, mask, ld, sw, Sbuf, out);
}